// GCN_16097537425900
// MI455X (gfx1250) — compile-verified
//
#include <hip/hip_runtime.h>

typedef __attribute__((ext_vector_type(2))) float v2f;
typedef __attribute__((ext_vector_type(8))) float v8f;

#define NNODES 100000
#define NEDGES 1600000
#define CIN 128
#define CH 128
#define COUT 64
#define NHIDL 3

// ---------------------------------------------------------------- degree
__global__ void deg_kernel(const int* __restrict__ dst, float* __restrict__ deg, int E) {
  int e = blockIdx.x * blockDim.x + threadIdx.x;
  if (e < E) atomicAdd(&deg[dst[e]], 1.0f);
}

__global__ void invdeg_kernel(float* __restrict__ deg, int n) {
  int i = blockIdx.x * blockDim.x + threadIdx.x;
  if (i < n) { float d = deg[i]; deg[i] = 1.0f / fmaxf(d, 1.0f); }
}

// ------------------------------------------------- scatter-add aggregation
// One wave per edge; lane handles 4 channels (float4 gather + 4 f32 atomics).
// h rows (51 MB total) are L2-resident; atomics resolve at L2.
__global__ __launch_bounds__(256)
void scatter_add_kernel(const float* __restrict__ h, const int* __restrict__ src,
                        const int* __restrict__ dst, float* __restrict__ s, int E) {
  unsigned tid = blockIdx.x * blockDim.x + threadIdx.x;
  unsigned e = tid >> 5;
  int lane = tid & 31;
  if (e >= (unsigned)E) return;
  int sv = src[e], dv = dst[e];
  const float4* hp = reinterpret_cast<const float4*>(h + (size_t)sv * CH);
  float4 v = hp[lane];
  float* sp = s + (size_t)dv * CH + lane * 4;
  atomicAdd(sp + 0, v.x);
  atomicAdd(sp + 1, v.y);
  atomicAdd(sp + 2, v.z);
  atomicAdd(sp + 3, v.w);
}

// ---------------------------------------------------- fused SAGE GEMM (WMMA)
// out[m, n] = (scale0[m]*A0[m,:]) @ W0[n,:]  (+ A1[m,:] @ W1[n,:]) + bias[n]
// epilogue: optional pre-activation copy (out2), ReLU, +0.2*skip.
// One 16x16 C-tile per wave via V_WMMA_F32_16X16X4_F32.
// A-frag 16x4: lane l -> row m0+(l&15), k = k0 + 2*(l>>4) + {0,1}  (float2)
// B-frag 4x16: lane l -> col n0+(l&15), same k pair; B[k,n] = W[n,k] (float2)
__global__ __launch_bounds__(256)
void sage_gemm_kernel(const float* __restrict__ A0, const float* __restrict__ scale0,
                      const float* __restrict__ A1,
                      const float* __restrict__ W0, const float* __restrict__ W1,
                      const float* __restrict__ bias,
                      const float* __restrict__ skip,
                      float* __restrict__ out, float* __restrict__ out2,
                      int M, int NOUT, int K0, int K1, int relu_flag) {
  const int wave = threadIdx.x >> 5;
  const int lane = threadIdx.x & 31;
  const int tiles_n = NOUT >> 4;                   // 8 (H) or 4 (OUT)
  const int mt = wave / tiles_n;
  const int nt = wave - mt * tiles_n;
  const int waves = blockDim.x >> 5;
  const int rows_per_block = 16 * (waves / tiles_n);
  const int m0 = blockIdx.x * rows_per_block + mt * 16;
  if (m0 >= M) return;
  const int n0 = nt << 4;
  const int lm = lane & 15;
  const int lk = (lane >> 4) << 1;                 // 0 or 2
  int row = m0 + lm;
  if (row >= M) row = M - 1;                       // clamp (stores are guarded)
  const int col = n0 + lm;

  v8f acc = {};

  {  // K0 block: aggregated neighbors (scaled by inv_deg)
    const float sc = scale0 ? scale0[row] : 1.0f;
    const float* ap = A0 + (size_t)row * K0 + lk;
    const float* bp = W0 + (size_t)col * K0 + lk;
#pragma unroll 8
    for (int k = 0; k < K0; k += 4) {
      v2f a = *reinterpret_cast<const v2f*>(ap + k);
      v2f b = *reinterpret_cast<const v2f*>(bp + k);
      a *= sc;
      acc = __builtin_amdgcn_wmma_f32_16x16x4_f32(false, a, false, b,
                                                  (short)0, acc, false, false);
    }
  }
  if (A1) {  // K1 block: root features
    const float* ap = A1 + (size_t)row * K1 + lk;
    const float* bp = W1 + (size_t)col * K1 + lk;
#pragma unroll 8
    for (int k = 0; k < K1; k += 4) {
      v2f a = *reinterpret_cast<const v2f*>(ap + k);
      v2f b = *reinterpret_cast<const v2f*>(bp + k);
      acc = __builtin_amdgcn_wmma_f32_16x16x4_f32(false, a, false, b,
                                                  (short)0, acc, false, false);
    }
  }

  // epilogue: C/D layout -> vgpr r, lane l holds (m0 + r + 8*(l>>4), n0 + (l&15))
  const float bv = bias ? bias[col] : 0.0f;
  const int mr = m0 + (lane >> 4) * 8;
#pragma unroll
  for (int r = 0; r < 8; ++r) {
    int rr = mr + r;
    if (rr >= M) continue;
    size_t idx = (size_t)rr * NOUT + col;
    float v = acc[r] + bv;
    if (out2) out2[idx] = v;                       // pre-activation (inp)
    if (relu_flag) v = fmaxf(v, 0.0f);
    if (skip) v += 0.2f * skip[idx];
    out[idx] = v;
  }
}

// ------------------------------------------------------------- log_softmax
__global__ void log_softmax_kernel(float* __restrict__ out, int M, int C) {
  int i = blockIdx.x * blockDim.x + threadIdx.x;
  if (i >= M) return;
  float* p = out + (size_t)i * C;
  float mx = -3.4028235e38f;
  for (int c = 0; c < C; ++c) mx = fmaxf(mx, p[c]);
  float sum = 0.0f;
  for (int c = 0; c < C; ++c) sum += expf(p[c] - mx);
  float lse = mx + logf(sum);
  for (int c = 0; c < C; ++c) p[c] -= lse;
}

// ---------------------------------------------------------------- launcher
extern "C" void kernel_launch(void* const* d_in, const int* in_sizes, int n_in,
                              void* d_out, int out_size, void* d_ws, size_t ws_size,
                              hipStream_t stream) {
  const float* x    = (const float*)d_in[0];
  const int*   eix  = (const int*)d_in[1];
  const float* Wp   = (const float*)d_in[2];
  const float* bp   = (const float*)d_in[3];
  const float* Wl_h = (const float*)d_in[4];
  const float* bl_h = (const float*)d_in[5];
  const float* Wr_h = (const float*)d_in[6];
  const float* Wl_o = (const float*)d_in[7];
  const float* bl_o = (const float*)d_in[8];
  const float* Wr_o = (const float*)d_in[9];
  const int* src = eix;
  const int* dst = eix + NEDGES;

  float* ws     = (float*)d_ws;
  float* invdeg = ws;                                  // N
  float* inp    = invdeg + NNODES;                     // N*CH
  float* hA     = inp + (size_t)NNODES * CH;           // N*CH
  float* hB     = hA  + (size_t)NNODES * CH;           // N*CH
  float* sbuf   = hB  + (size_t)NNODES * CH;           // N*CH

  // degree -> inv_deg (in place)
  hipMemsetAsync(invdeg, 0, NNODES * sizeof(float), stream);
  deg_kernel<<<(NEDGES + 255) / 256, 256, 0, stream>>>(dst, invdeg, NEDGES);
  invdeg_kernel<<<(NNODES + 255) / 256, 256, 0, stream>>>(invdeg, NNODES);

  // inProj: inp = x@Wp.T + bp ; hA = relu(inp)
  sage_gemm_kernel<<<(NNODES + 15) / 16, 256, 0, stream>>>(
      x, nullptr, nullptr, Wp, nullptr, bp, nullptr, hA, inp,
      NNODES, CH, CIN, 0, 1);

  const int scat_blocks = (NEDGES * 32 + 255) / 256;
  float* hcur = hA;
  float* hnext = hB;
  for (int i = 0; i < NHIDL; ++i) {
    hipMemsetAsync(sbuf, 0, (size_t)NNODES * CH * sizeof(float), stream);
    scatter_add_kernel<<<scat_blocks, 256, 0, stream>>>(hcur, src, dst, sbuf, NEDGES);
    sage_gemm_kernel<<<(NNODES + 15) / 16, 256, 0, stream>>>(
        sbuf, invdeg, hcur,
        Wl_h + (size_t)i * CH * CH, Wr_h + (size_t)i * CH * CH,
        bl_h + (size_t)i * CH, inp, hnext, nullptr,
        NNODES, CH, CH, CH, 1);
    float* t = hcur; hcur = hnext; hnext = t;
  }

  // output SAGE layer -> d_out, then in-place log_softmax
  hipMemsetAsync(sbuf, 0, (size_t)NNODES * CH * sizeof(float), stream);
  scatter_add_kernel<<<scat_blocks, 256, 0, stream>>>(hcur, src, dst, sbuf, NEDGES);
  sage_gemm_kernel<<<(NNODES + 31) / 32, 256, 0, stream>>>(
      sbuf, invdeg, hcur, Wl_o, Wr_o, bl_o, nullptr, (float*)d_out, nullptr,
      NNODES, COUT, CH, CH, 0);
  log_softmax_kernel<<<(NNODES + 255) / 256, 256, 0, stream>>>(
      (float*)d_out, NNODES, COUT);
}